// GPT2_57853209477607
// MI455X (gfx1250) — compile-verified
//
#include <hip/hip_runtime.h>
#include <hip/hip_bf16.h>
#include <math.h>

// GPT-2 forward for MI455X (gfx1250). All GEMMs on the WMMA pipe with exact
// f32 arithmetic (v_wmma_f32_16x16x4_f32). Workspace requirement:
//   (3*4096*768 + 4096*3072) * 4 bytes = 88,080,384 bytes (~84 MB).

typedef float v2f __attribute__((ext_vector_type(2)));
typedef float v8f __attribute__((ext_vector_type(8)));

constexpr int kD   = 768;
constexpr int kH   = 12;
constexpr int kL   = 12;
constexpr int kS   = 1024;
constexpr int kB   = 4;
constexpr int kFF  = 3072;
constexpr int kHD  = 64;
constexpr int kTok = kB * kS;        // 4096 rows

// ---------------------------------------------------------------- embedding
__global__ __launch_bounds__(256)
void embed_kernel(const int* __restrict__ tokens, const float* __restrict__ wte,
                  const float* __restrict__ wpe, float* __restrict__ x) {
    int row = blockIdx.x;                 // b*S + s
    int s   = row % kS;
    int tok = tokens[row];
    const float* te = wte + (size_t)tok * kD;
    const float* pe = wpe + (size_t)s * kD;
    float* o = x + (size_t)row * kD;
    for (int i = threadIdx.x; i < kD; i += 256) o[i] = te[i] + pe[i];
}

// ---------------------------------------------------------------- layernorm
__device__ __forceinline__ float block_sum(float v, float* red) {
    int tid = threadIdx.x;
    red[tid] = v; __syncthreads();
    for (int s = 128; s > 0; s >>= 1) {
        if (tid < s) red[tid] += red[tid + s];
        __syncthreads();
    }
    float r = red[0]; __syncthreads();
    return r;
}

// out = LN(in + resid) * w + b ; resid may be null; in-place (out==in) safe.
__global__ __launch_bounds__(256)
void ln_kernel(const float* in, const float* resid,
               const float* w, const float* b, float* out) {
    __shared__ float red[256];
    int row = blockIdx.x, tid = threadIdx.x;
    const float* ip = in + (size_t)row * kD;
    const float* rp = resid ? resid + (size_t)row * kD : nullptr;
    float v[3];
    float s = 0.f;
    #pragma unroll
    for (int k = 0; k < 3; ++k) {
        int i = tid + 256 * k;
        v[k] = ip[i] + (rp ? rp[i] : 0.f);
        s += v[k];
    }
    float mean = block_sum(s, red) * (1.f / kD);
    float q = 0.f;
    #pragma unroll
    for (int k = 0; k < 3; ++k) { float d = v[k] - mean; q += d * d; }
    float var = block_sum(q, red) * (1.f / kD);
    float rs = rsqrtf(var + 1e-5f);
    float* op = out + (size_t)row * kD;
    #pragma unroll
    for (int k = 0; k < 3; ++k) {
        int i = tid + 256 * k;
        op[i] = (v[k] - mean) * rs * w[i] + b[i];
    }
}

// ---------------------------------------------------------------- WMMA GEMM
// C[M,N] = A[M,K] @ W[K,N] + bias ; mode 0: plain, 1: +GELU(exact), 2: +resid
// Requires M%64==0, N%128==0, K%32==0 (true for all GPT-2 shapes here).
//
// B tile is stored K-pair-interleaved in LDS: element (k,n) lives at
// Bs[(k>>1)*BSTP + 2*n + (k&1)], so a WMMA B-fragment (k, k+1 at fixed n,
// k even) is one aligned 8-byte pair -> single ds_load_b64, no packing movs.
// BSTP % 64 == 32 puts the two half-waves' pair-rows on disjoint bank halves.
constexpr int BM = 64, BN = 128, BK = 32;
constexpr int AST  = 34;            // padded A row stride (even -> aligned pairs)
constexpr int BSTP = 2 * BN + 32;   // 288 dwords per k-pair row

__global__ __launch_bounds__(256)
void gemm_wmma_kernel(const float* __restrict__ A, const float* __restrict__ W,
                      const float* __restrict__ bias, const float* __restrict__ resid,
                      float* __restrict__ C, int M, int N, int K, int mode) {
    __shared__ float As[BM * AST];            //  8,704 B
    __shared__ float Bs[(BK / 2) * BSTP];     // 18,432 B
    const int tid  = threadIdx.x;
    const int lane = tid & 31;
    const int hi   = lane >> 4;       // half-wave: selects K pair
    const int l16  = lane & 15;
    const int wave = tid >> 5;        // 8 waves
    const int wm   = wave >> 1;       // 0..3 : 16-row strip
    const int wn   = wave & 1;        // 0..1 : 64-col strip
    const int tileN = blockIdx.x * BN;
    const int tileM = blockIdx.y * BM;

    v8f acc[4] = {};

    for (int k0 = 0; k0 < K; k0 += BK) {
        // stage A tile (64x32), coalesced
        for (int i = tid; i < BM * BK; i += 256) {
            int r = i >> 5, c = i & 31;
            As[r * AST + c] = A[(size_t)(tileM + r) * K + k0 + c];
        }
        // stage B tile (32x128), coalesced reads, pair-interleaved writes
        for (int i = tid; i < BK * BN; i += 256) {
            int r = i >> 7, c = i & 127;
            Bs[(r >> 1) * BSTP + 2 * c + (r & 1)] = W[(size_t)(k0 + r) * N + tileN + c];
        }
        __syncthreads();

        #pragma unroll
        for (int kk = 0; kk < BK; kk += 4) {
            const int ka = kk + 2 * hi;       // even K index for this half-wave
            const v2f a = *(const v2f*)&As[(wm * 16 + l16) * AST + ka];
            const float* bsrow = &Bs[(ka >> 1) * BSTP];
            #pragma unroll
            for (int j = 0; j < 4; ++j) {
                const int bcol = wn * 64 + j * 16 + l16;
                const v2f bb = *(const v2f*)&bsrow[2 * bcol];
                acc[j] = __builtin_amdgcn_wmma_f32_16x16x4_f32(
                    false, a, false, bb, (short)0, acc[j], false, false);
            }
        }
        __syncthreads();
    }

    // epilogue: D layout — VGPR r holds row r (lanes 0-15) / row r+8 (lanes 16-31)
    #pragma unroll
    for (int j = 0; j < 4; ++j) {
        const int col = tileN + wn * 64 + j * 16 + l16;
        const float bv = bias[col];
        #pragma unroll
        for (int r = 0; r < 8; ++r) {
            const int row = tileM + wm * 16 + 8 * hi + r;
            float vv = acc[j][r] + bv;
            if (mode == 1)       vv = 0.5f * vv * (1.f + erff(vv * 0.70710678118654752f));
            else if (mode == 2)  vv += resid[(size_t)row * N + col];
            C[(size_t)row * N + col] = vv;
        }
    }
}

// ------------------------------------------------------- K/V -> presents copy
__global__ __launch_bounds__(256)
void copykv_kernel(const float* __restrict__ qkv,
                   float* __restrict__ outk, float* __restrict__ outv) {
    size_t i = (size_t)blockIdx.x * 256 + threadIdx.x;   // over Tok*D
    size_t row = i / kD, c = i % kD;
    const float* src = qkv + row * (size_t)(3 * kD);
    outk[i] = src[kD + c];
    outv[i] = src[2 * kD + c];
}

// ------------------------------------------------------------------ attention
// one block per (query q, head bh). Mask applied before scale in reference:
// masked entries = -1e20 -> exp underflows to exactly 0, so only j<=q matter.
__global__ __launch_bounds__(256)
void attn_kernel(const float* __restrict__ qkv, float* __restrict__ aout) {
    __shared__ float sc[kS];
    __shared__ float qv[kHD];
    __shared__ float red[256];
    const int q  = blockIdx.x;
    const int bh = blockIdx.y;
    const int b = bh / kH, h = bh % kH;
    const int tid = threadIdx.x;
    const size_t base = (size_t)b * kS * (3 * kD);
    const float* qrow = qkv + base + (size_t)q * (3 * kD) + h * kHD;
    if (tid < kHD) qv[tid] = qrow[tid];
    __syncthreads();

    // scaled scores for valid keys
    for (int j = tid; j <= q; j += 256) {
        const float* krow = qkv + base + (size_t)j * (3 * kD) + kD + h * kHD;
        float d = 0.f;
        #pragma unroll
        for (int c = 0; c < kHD; ++c) d += qv[c] * krow[c];
        sc[j] = d * 0.125f;                       // 1/sqrt(64)
    }
    __syncthreads();

    // max
    float lm = -3.4e38f;
    for (int j = tid; j <= q; j += 256) lm = fmaxf(lm, sc[j]);
    red[tid] = lm; __syncthreads();
    for (int s = 128; s > 0; s >>= 1) {
        if (tid < s) red[tid] = fmaxf(red[tid], red[tid + s]);
        __syncthreads();
    }
    const float m = red[0]; __syncthreads();

    // exp + sum
    float ls = 0.f;
    for (int j = tid; j <= q; j += 256) { float e = expf(sc[j] - m); sc[j] = e; ls += e; }
    __syncthreads();
    red[tid] = ls; __syncthreads();
    for (int s = 128; s > 0; s >>= 1) {
        if (tid < s) red[tid] += red[tid + s];
        __syncthreads();
    }
    const float inv = 1.f / red[0];
    __syncthreads();

    // attn @ V : 4 j-groups x 64 dims (coalesced V reads)
    const int jg = tid >> 6;
    const int d  = tid & 63;
    const float* vbase = qkv + base + 2 * kD + h * kHD;
    float acc = 0.f;
    for (int j = jg; j <= q; j += 4) acc += sc[j] * vbase[(size_t)j * (3 * kD) + d];
    red[tid] = acc; __syncthreads();
    if (tid < 64) {
        float o = (red[tid] + red[tid + 64] + red[tid + 128] + red[tid + 192]) * inv;
        aout[((size_t)b * kS + q) * kD + h * kHD + d] = o;
    }
}

// ------------------------------------------------------------------- launcher
extern "C" void kernel_launch(void* const* d_in, const int* in_sizes, int n_in,
                              void* d_out, int out_size, void* d_ws, size_t ws_size,
                              hipStream_t stream) {
    (void)in_sizes; (void)n_in; (void)out_size; (void)ws_size;
    const int*   tokens = (const int*)  d_in[0];
    const float* wte  = (const float*)d_in[1];
    const float* wpe  = (const float*)d_in[2];
    const float* ln1w = (const float*)d_in[3];
    const float* ln1b = (const float*)d_in[4];
    const float* ln2w = (const float*)d_in[5];
    const float* ln2b = (const float*)d_in[6];
    const float* Wexp = (const float*)d_in[7];
    const float* bexp = (const float*)d_in[8];
    const float* Wo   = (const float*)d_in[9];
    const float* bo   = (const float*)d_in[10];
    const float* Wf1  = (const float*)d_in[11];
    const float* bf1  = (const float*)d_in[12];
    const float* Wf2  = (const float*)d_in[13];
    const float* bf2  = (const float*)d_in[14];
    float* out = (float*)d_out;

    const size_t nXD = (size_t)kTok * kD;            // 3,145,728 floats
    float* x  = (float*)d_ws;                        // residual stream
    float* ab = x  + nXD;                            // ln1-out, then attn-out
    float* pb = ab + nXD;                            // proj-out, then "add"
    float* qf = pb + nXD;                            // qkv (36MB) / ff (48MB) union

    embed_kernel<<<kTok, 256, 0, stream>>>(tokens, wte, wpe, x);

    for (int l = 0; l < kL; ++l) {
        // h = LN1(x)
        ln_kernel<<<kTok, 256, 0, stream>>>(x, nullptr, ln1w + l * kD, ln1b + l * kD, ab);
        // qkv = h @ W_exp + b_exp
        gemm_wmma_kernel<<<dim3(3 * kD / BN, kTok / BM), 256, 0, stream>>>(
            ab, Wexp + (size_t)l * kD * 3 * kD, bexp + (size_t)l * 3 * kD,
            nullptr, qf, kTok, 3 * kD, kD, 0);
        // presents[l] = {k, v}
        float* outk = out + nXD + (size_t)l * 2 * nXD;
        copykv_kernel<<<(unsigned)(nXD / 256), 256, 0, stream>>>(qf, outk, outk + nXD);
        // a = attention(qkv)
        attn_kernel<<<dim3(kS, kB * kH), 256, 0, stream>>>(qf, ab);
        // proj = a @ W_o + b_o
        gemm_wmma_kernel<<<dim3(kD / BN, kTok / BM), 256, 0, stream>>>(
            ab, Wo + (size_t)l * kD * kD, bo + (size_t)l * kD,
            nullptr, pb, kTok, kD, kD, 0);
        // add = LN2(proj + x)   (in-place on pb)
        ln_kernel<<<kTok, 256, 0, stream>>>(pb, x, ln2w + l * kD, ln2b + l * kD, pb);
        // ff = gelu(add @ W_ff1 + b_ff1)
        gemm_wmma_kernel<<<dim3(kFF / BN, kTok / BM), 256, 0, stream>>>(
            pb, Wf1 + (size_t)l * kD * kFF, bf1 + (size_t)l * kFF,
            nullptr, qf, kTok, kFF, kD, 1);
        // x = ff @ W_ff2 + b_ff2 + add
        gemm_wmma_kernel<<<dim3(kD / BN, kTok / BM), 256, 0, stream>>>(
            qf, Wf2 + (size_t)l * kFF * kD, bf2 + (size_t)l * kD,
            pb, x, kTok, kD, kFF, 2);
    }

    hipMemcpyAsync(d_out, x, nXD * sizeof(float), hipMemcpyDeviceToDevice, stream);
}